// BraLM_22479858827801
// MI455X (gfx1250) — compile-verified
//
#include <hip/hip_runtime.h>
#include <hip/hip_bf16.h>
#include <math.h>

// Problem dims (from reference)
#define Hdim 64
#define Bdim 16
#define Tdim 64
#define Kdim 32
#define SEQdim 512

typedef float v2f __attribute__((ext_vector_type(2)));
typedef float v8f __attribute__((ext_vector_type(8)));
typedef int   v4i_vs __attribute__((vector_size(4 * sizeof(int))));   // b128 payload type
typedef __attribute__((address_space(1))) v4i_vs* gbl_v4i_ptr;        // global (AS1)
typedef __attribute__((address_space(3))) v4i_vs* lds_v4i_ptr;        // LDS (AS3)

// ---- feature probes (both confirmed present by earlier rounds) ------------
#if defined(__has_builtin)
#if __has_builtin(__builtin_amdgcn_wmma_f32_16x16x4_f32)
#define HAVE_WMMA_F32X4 1
#endif
#if __has_builtin(__builtin_amdgcn_global_load_async_to_lds_b128)
#define HAVE_ASYNC_LDS 1
#endif
#endif
#ifndef HAVE_WMMA_F32X4
#define HAVE_WMMA_F32X4 0
#endif
#ifndef HAVE_ASYNC_LDS
#define HAVE_ASYNC_LDS 0
#endif

__device__ __forceinline__ float gelu_exact(float x) {
    return 0.5f * x * (1.0f + erff(x * 0.70710678118654752f));
}

// ---------------------------------------------------------------------------
// init: positional encoding table, e0 = gelu(1/H + node_bias[nid] + pe[0]),
// zero the cache. pe[0] is sin(0)/cos(0) = 0/1, computed inline (no race).
// ---------------------------------------------------------------------------
__global__ void bralm_init_kernel(float* __restrict__ pe,
                                  float* __restrict__ cache,
                                  float* __restrict__ e0,
                                  const float* __restrict__ node_bias,
                                  const int* __restrict__ neighbor_ids) {
    int gid = blockIdx.x * blockDim.x + threadIdx.x;   // 65536 threads
    if (gid < SEQdim * Hdim) {
        int s = gid >> 6, h = gid & 63;
        // ang = s * 10000^( (2*(h/2)) / H )   (reference multiplies, not divides)
        float ang = (float)s * powf(10000.0f, (float)(h & ~1) / (float)Hdim);
        pe[gid] = (h & 1) ? cosf(ang) : sinf(ang);
    }
    if (gid < Bdim * Tdim * Hdim) cache[gid] = 0.0f;
    if (gid < Bdim * Hdim) {
        int b = gid >> 6, h = gid & 63;
        int nid = neighbor_ids[b * Tdim * Kdim * 2];   // neighbor_ids[b,0,0,0]
        float pe0 = (h & 1) ? 1.0f : 0.0f;
        float x = 1.0f / (float)Hdim + node_bias[nid * Hdim + h] + pe0;
        e0[gid] = gelu_exact(x);
    }
}

// ---------------------------------------------------------------------------
// prep (per step i): attn = softmax(mask ? pos : -1e9) over T, then
// et[b,h] = sum_t attn[t]*cache[b,t,h]   (et := e0 when i==0).
// Masked entries underflow to exactly 0 in f32 (same as the reference), so
// both the softmax denominator and the contraction stop at t < step.
// One block per b (16 blocks) -> coalesced 16KB slice per block instead of a
// single workgroup streaming the whole 256KB cache.
// ---------------------------------------------------------------------------
__global__ void bralm_prep_kernel(const float* __restrict__ positions,
                                  const float* __restrict__ e0,
                                  const float* __restrict__ cache,
                                  float* __restrict__ et,
                                  int step) {
    __shared__ float attn[Tdim];
    const int b = blockIdx.x;        // 0..15
    const int h = threadIdx.x;       // 0..63
    if (step == 0) {
        et[b * Hdim + h] = e0[b * Hdim + h];
        return;
    }
    if (h == 0) {
        float m = -INFINITY;
        for (int t = 0; t < step; ++t) m = fmaxf(m, positions[t]);
        float s = 0.0f;
        for (int t = 0; t < step; ++t) s += expf(positions[t] - m);
        float inv = 1.0f / s;
        for (int t = 0; t < step; ++t) attn[t] = expf(positions[t] - m) * inv;
    }
    __syncthreads();
    float v = 0.0f;
    const float* crow = cache + (size_t)b * Tdim * Hdim + h;
    for (int t = 0; t < step; ++t) v += attn[t] * crow[t * Hdim];
    et[b * Hdim + h] = v;
}

// ---------------------------------------------------------------------------
// main (per step i): one block per (b,k). Stage W[idx] (16 KB f32) into LDS
// with GLOBAL_LOAD_ASYNC_TO_LDS_B128 (coalesced 16B/lane), then 4 waves each
// produce a 16-wide slice of y = et @ W via V_WMMA_F32_16X16X4_F32 (A row 0
// holds et, rows 1..15 forced to zero with a hoisted lane-mask multiply).
// Epilogue: +bias +pe[i+1], gelu, ||y|| via shfl-xor tree; k==0 writes cache.
// ---------------------------------------------------------------------------
__global__ void bralm_main_kernel(const int* __restrict__ param_indices,
                                  const float* __restrict__ weights,
                                  const float* __restrict__ biases,
                                  const float* __restrict__ pe,
                                  const float* __restrict__ et,
                                  float* __restrict__ cache,
                                  float* __restrict__ energy,
                                  int step) {
    __shared__ float Wl[Hdim * Hdim];   // 16 KB
    __shared__ float etl[Hdim];
    __shared__ float red[4];            // one partial per wave

    const int tid  = threadIdx.x;        // 128 threads = 4 waves
    const int lane = tid & 31;
    const int wave = tid >> 5;
    const int b    = blockIdx.x >> 5;    // blockIdx = b*K + k
    const int k    = blockIdx.x & 31;

    const int idx = param_indices[(b * Tdim + step) * Kdim + k];
    const float* Wg = weights + (size_t)idx * (Hdim * Hdim);

    // ---- stage W into LDS, 16 B per lane per issue (fully coalesced) ------
#if HAVE_ASYNC_LDS
    for (int it = 0; it < 8; ++it) {
        int o = (it * 128 + tid) * 4;    // float index, 16B granules
        __builtin_amdgcn_global_load_async_to_lds_b128(
            (gbl_v4i_ptr)(Wg + o),
            (lds_v4i_ptr)(Wl + o),
            0, 0);
    }
    // overlap the tiny et staging with the in-flight DMA
    if (tid < Hdim) etl[tid] = et[b * Hdim + tid];
    asm volatile("s_wait_asynccnt 0" ::: "memory");
#else
    for (int it = 0; it < 8; ++it) {
        int o = it * 128 + tid;          // float4 index (1024 total)
        ((float4*)Wl)[o] = ((const float4*)Wg)[o];
    }
    if (tid < Hdim) etl[tid] = et[b * Hdim + tid];
#endif
    __syncthreads();

    const int d0   = wave * 16;          // this wave's output slice
    const int half = lane >> 4;          // 0 or 1 (K=0/1 vs K=2/3 rows)
    const int col  = lane & 15;
    // Hoisted lane mask: only row M==0 of A is real -> one v_cndmask outside
    // the loop, packed v_pk_mul inside (keeps EXEC untouched around WMMAs).
    const float amask = (col == 0) ? 1.0f : 0.0f;
    const int   arow  = 2 * half;        // et element pair base within chunk

    float y;
#if HAVE_WMMA_F32X4
    v8f c = {0.f, 0.f, 0.f, 0.f, 0.f, 0.f, 0.f, 0.f};
#pragma unroll
    for (int h0 = 0; h0 < Hdim; h0 += 4) {
        // A (16x4 f32): VGPR0 lanes 0-15 K=0 / lanes 16-31 K=2; VGPR1 K=1/K=3.
        v2f a, bb;
        a.x = amask * etl[h0 + arow + 0];
        a.y = amask * etl[h0 + arow + 1];
        // B (4x16 f32): VGPR0 rows K=0/K=2 striped over N; VGPR1 rows K=1/K=3.
        bb.x = Wl[(h0 + arow + 0) * Hdim + d0 + col];
        bb.y = Wl[(h0 + arow + 1) * Hdim + d0 + col];
        c = __builtin_amdgcn_wmma_f32_16x16x4_f32(
                false, a, false, bb, (short)0, c, false, false);
    }
    // C/D layout: VGPR0 lanes 0-15 = (M=0, N=lane) -> our y slice
    y = c[0];
#else
    // scalar fallback (keeps file compiling if the builtin is absent)
    y = 0.0f;
    if (lane < 16) {
        int d = d0 + lane;
        for (int h = 0; h < Hdim; ++h) y += etl[h] * Wl[h * Hdim + d];
    }
#endif

    // ---- epilogue ---------------------------------------------------------
    float sq = 0.0f;
    if (lane < 16) {
        int d = d0 + lane;
        float x = y + biases[(size_t)idx * Hdim + d] + pe[(step + 1) * Hdim + d];
        float g = gelu_exact(x);
        sq = g * g;
        if (k == 0) cache[((size_t)b * Tdim + step) * Hdim + d] = g;
    }
    // reduce lanes 0..15 within the wave (xor offsets stay inside the half)
    sq += __shfl_xor(sq, 1, 32);
    sq += __shfl_xor(sq, 2, 32);
    sq += __shfl_xor(sq, 4, 32);
    sq += __shfl_xor(sq, 8, 32);
    if (lane == 0) red[wave] = sq;
    __syncthreads();
    if (tid == 0) {
        float s = red[0] + red[1] + red[2] + red[3];
        energy[((size_t)step * Bdim + b) * Kdim + k] = sqrtf(s);
    }
}

// ---------------------------------------------------------------------------
// loss: (1/T) * sum_i mean_b( logsumexp_k(energy[i,b,:]) - energy[i,b,0] )
// ---------------------------------------------------------------------------
__global__ void bralm_loss_kernel(const float* __restrict__ energy,
                                  float* __restrict__ out) {
    __shared__ float red[256];
    int tid = threadIdx.x;
    float acc = 0.0f;
    for (int p = tid; p < Tdim * Bdim; p += blockDim.x) {
        const float* e = energy + (size_t)p * Kdim;
        float m = e[0];
        for (int j = 1; j < Kdim; ++j) m = fmaxf(m, e[j]);
        float s = 0.0f;
        for (int j = 0; j < Kdim; ++j) s += expf(e[j] - m);
        acc += (m + logf(s)) - e[0];
    }
    red[tid] = acc;
    __syncthreads();
    if (tid == 0) {
        float s = 0.0f;
        for (int j = 0; j < 256; ++j) s += red[j];
        out[0] = s / (float)(Bdim * Tdim);
    }
}

// ---------------------------------------------------------------------------
extern "C" void kernel_launch(void* const* d_in, const int* in_sizes, int n_in,
                              void* d_out, int out_size, void* d_ws, size_t ws_size,
                              hipStream_t stream) {
    const int*   neighbor_ids  = (const int*)  d_in[0];   // (B,T,K,2)
    const int*   param_indices = (const int*)  d_in[1];   // (B,T,K)
    const float* weights       = (const float*)d_in[2];   // (E,H,H)
    const float* biases        = (const float*)d_in[3];   // (E,1,H)
    const float* node_bias     = (const float*)d_in[4];   // (N,1,H)
    const float* positions     = (const float*)d_in[5];   // (1,SEQ,1)

    // workspace layout (floats), total ~533 KB
    float* ws     = (float*)d_ws;
    float* pe     = ws;                                   // SEQ*H   = 32768
    float* cache  = pe     + SEQdim * Hdim;               // B*T*H   = 65536
    float* e0     = cache  + Bdim * Tdim * Hdim;          // B*H     = 1024
    float* et     = e0     + Bdim * Hdim;                 // B*H     = 1024
    float* energy = et     + Bdim * Hdim;                 // T*B*K   = 32768

    bralm_init_kernel<<<256, 256, 0, stream>>>(pe, cache, e0, node_bias, neighbor_ids);
    for (int i = 0; i < Tdim; ++i) {
        bralm_prep_kernel<<<Bdim, Hdim, 0, stream>>>(positions, e0, cache, et, i);
        bralm_main_kernel<<<Bdim * Kdim, 128, 0, stream>>>(
            param_indices, weights, biases, pe, et, cache, energy, i);
    }
    bralm_loss_kernel<<<1, 256, 0, stream>>>(energy, (float*)d_out);
}